// TurtleAttnBlock_33011118637115
// MI455X (gfx1250) — compile-verified
//
#include <hip/hip_runtime.h>
#include <hip/hip_bf16.h>

typedef __attribute__((ext_vector_type(2))) float v2f;
typedef __attribute__((ext_vector_type(8))) float v8f;

#define BATCH 2
#define CCH   192
#define C3    576
#define HEADS 6
#define HD    32
#define HEIGHT 256
#define WIDTH  256
#define HW    65536
#define NCHUNK 128      // split-K chunks over HW for q.k^T
#define CHUNKLEN 512    // HW / NCHUNK

// ---------------------------------------------------------------------------
// Generic GEMM: Y[b,m,n] = sum_k W[m,k] * X[b,k,n]   (M mult of 32, K mult of 4,
// N mult of 512). One wave computes a 32(M) x 64(N) tile via
// V_WMMA_F32_16X16X4_F32: 2 A-fragments share 4 B-fragments -> 8 WMMA per
// 10 vmem ops. block = 256 threads = 8 waves along N.
// ---------------------------------------------------------------------------
__global__ __launch_bounds__(256) void gemm32_wmma(
    const float* __restrict__ W, const float* __restrict__ X,
    float* __restrict__ Y, int M, int K, int N) {
  const int lane = threadIdx.x & 31;
  const int wave = threadIdx.x >> 5;
  const int l16  = lane & 15;
  const int half = lane >> 4;            // 0 -> K pair (0,1), 1 -> (2,3)
  const int b  = blockIdx.z;
  const int m0 = blockIdx.y * 32;
  const int n0 = (blockIdx.x * 8 + wave) * 64;
  const float* Xb = X + (size_t)b * K * N;
  float* Yb       = Y + (size_t)b * M * N;

  v8f accA0 = {}, accA1 = {}, accA2 = {}, accA3 = {};   // rows m0..m0+15
  v8f accB0 = {}, accB1 = {}, accB2 = {}, accB3 = {};   // rows m0+16..m0+31
  const float* wrow0 = W + (size_t)(m0 + l16) * K + 2 * half;
  const float* wrow1 = wrow0 + (size_t)16 * K;
  for (int k = 0; k < K; k += 4) {
    v2f a0; a0.x = wrow0[k]; a0.y = wrow0[k + 1];
    v2f a1; a1.x = wrow1[k]; a1.y = wrow1[k + 1];
    const float* xr = Xb + (size_t)(k + 2 * half) * N + n0 + l16;
    // prefetch the X rows needed 2 k-steps ahead while WMMAs run
    __builtin_prefetch(xr + (size_t)8 * N, 0, 0);
    __builtin_prefetch(xr + (size_t)9 * N, 0, 0);
    v2f b0; b0.x = xr[0];  b0.y = xr[(size_t)N + 0];
    v2f b1; b1.x = xr[16]; b1.y = xr[(size_t)N + 16];
    v2f b2; b2.x = xr[32]; b2.y = xr[(size_t)N + 32];
    v2f b3; b3.x = xr[48]; b3.y = xr[(size_t)N + 48];
    accA0 = __builtin_amdgcn_wmma_f32_16x16x4_f32(false, a0, false, b0, (short)0, accA0, false, false);
    accA1 = __builtin_amdgcn_wmma_f32_16x16x4_f32(false, a0, false, b1, (short)0, accA1, false, false);
    accA2 = __builtin_amdgcn_wmma_f32_16x16x4_f32(false, a0, false, b2, (short)0, accA2, false, false);
    accA3 = __builtin_amdgcn_wmma_f32_16x16x4_f32(false, a0, false, b3, (short)0, accA3, false, false);
    accB0 = __builtin_amdgcn_wmma_f32_16x16x4_f32(false, a1, false, b0, (short)0, accB0, false, false);
    accB1 = __builtin_amdgcn_wmma_f32_16x16x4_f32(false, a1, false, b1, (short)0, accB1, false, false);
    accB2 = __builtin_amdgcn_wmma_f32_16x16x4_f32(false, a1, false, b2, (short)0, accB2, false, false);
    accB3 = __builtin_amdgcn_wmma_f32_16x16x4_f32(false, a1, false, b3, (short)0, accB3, false, false);
  }
  // C/D layout: lanes 0-15: N=lane, M=v; lanes 16-31: N=lane-16, M=v+8
  float* y0 = Yb + (size_t)(m0 + half * 8) * N + n0 + l16;
  float* y1 = y0 + (size_t)16 * N;
#pragma unroll
  for (int v = 0; v < 8; ++v) {
    y0[(size_t)v * N + 0]  = accA0[v];
    y0[(size_t)v * N + 16] = accA1[v];
    y0[(size_t)v * N + 32] = accA2[v];
    y0[(size_t)v * N + 48] = accA3[v];
    y1[(size_t)v * N + 0]  = accB0[v];
    y1[(size_t)v * N + 16] = accB1[v];
    y1[(size_t)v * N + 32] = accB2[v];
    y1[(size_t)v * N + 48] = accB3[v];
  }
}

// ---------------------------------------------------------------------------
// Depthwise 3x3, SAME padding; splits channels into q (ws), k (d_out raw), v (d_out final)
// ---------------------------------------------------------------------------
__global__ __launch_bounds__(256) void dwconv3x3_split(
    const float* __restrict__ in, const float* __restrict__ wdw,
    float* __restrict__ oq, float* __restrict__ ok, float* __restrict__ ov) {
  size_t idx = (size_t)blockIdx.x * 256 + threadIdx.x;   // < 2*576*65536
  const int x  = (int)(idx & 255);
  const int y  = (int)((idx >> 8) & 255);
  const int pl = (int)(idx >> 16);       // plane index = b*576 + ch
  const int ch = pl % C3;
  const int b  = pl / C3;
  const float* wp = wdw + ch * 9;
  const float* ip = in + ((size_t)pl << 16);
  float acc = 0.f;
#pragma unroll
  for (int dy = -1; dy <= 1; ++dy) {
    int yy = y + dy;
    if ((unsigned)yy >= (unsigned)HEIGHT) continue;
#pragma unroll
    for (int dx = -1; dx <= 1; ++dx) {
      int xx = x + dx;
      if ((unsigned)xx >= (unsigned)WIDTH) continue;
      acc += wp[(dy + 1) * 3 + (dx + 1)] * ip[yy * WIDTH + xx];
    }
  }
  const size_t sp = (size_t)y * WIDTH + x;
  if (ch < CCH)            oq[(((size_t)b * CCH + ch) << 16) + sp] = acc;
  else if (ch < 2 * CCH)   ok[(((size_t)b * CCH + (ch - CCH)) << 16) + sp] = acc;
  else                     ov[(((size_t)b * CCH + (ch - 2 * CCH)) << 16) + sp] = acc;
}

// ---------------------------------------------------------------------------
// Per-row 1/max(||row||,eps) over HW for q (rows 0..383) and k (rows 384..767)
// ---------------------------------------------------------------------------
__global__ __launch_bounds__(256) void rownorm(
    const float* __restrict__ q, const float* __restrict__ k, float* __restrict__ rinv) {
  const int r = blockIdx.x;   // 0..767
  const float* row = (r < BATCH * CCH) ? q + ((size_t)r << 16)
                                       : k + ((size_t)(r - BATCH * CCH) << 16);
  float s = 0.f;
  for (int i = threadIdx.x; i < HW; i += 256) { float v = row[i]; s += v * v; }
  __shared__ float red[8];
#pragma unroll
  for (int off = 16; off; off >>= 1) s += __shfl_down(s, off, 32);
  if ((threadIdx.x & 31) == 0) red[threadIdx.x >> 5] = s;
  __syncthreads();
  if (threadIdx.x == 0) {
    float t = 0.f;
#pragma unroll
    for (int i = 0; i < 8; ++i) t += red[i];
    rinv[r] = 1.0f / fmaxf(sqrtf(t), 1e-12f);
  }
}

__global__ __launch_bounds__(256) void scale_rows(
    float* __restrict__ q, float* __restrict__ k, const float* __restrict__ rinv) {
  size_t idx = (size_t)blockIdx.x * 256 + threadIdx.x;   // < 2 * (2*192*65536)
  const size_t NQ = (size_t)BATCH * CCH * HW;
  if (idx < NQ) q[idx] *= rinv[idx >> 16];
  else { size_t j = idx - NQ; k[j] *= rinv[BATCH * CCH + (j >> 16)]; }
}

// ---------------------------------------------------------------------------
// Split-K attention logits: S[bh,c,d] = sum_n q[c,n]*k[d,n]; partials per chunk.
// grid = (16, 8, 12); wave -> (bh, mt, nt, chunk) 16x16 tile over 512-long chunk.
// ---------------------------------------------------------------------------
__global__ __launch_bounds__(256) void attn_qk_partial(
    const float* __restrict__ Q, const float* __restrict__ Kc,
    const float* __restrict__ Kn, float* __restrict__ P) {
  const int lane = threadIdx.x & 31;
  const int wave = threadIdx.x >> 5;
  const int l16  = lane & 15;
  const int half = lane >> 4;
  const int bh = blockIdx.z;                 // 0..11
  const int mt = blockIdx.y >> 2;            // 0..1
  const int nt = blockIdx.y & 3;             // 0..3
  const int chunk = blockIdx.x * 8 + wave;   // 0..127
  const int b = bh / HEADS, h = bh % HEADS;
  const float* qrow = Q + (((size_t)b * CCH + h * HD + mt * 16 + l16) << 16);
  const int d = nt * 16 + l16;               // 0..63 over concat(kc, kn)
  const float* krow = (d < HD)
      ? Kc + (((size_t)bh * HD + d) << 16)
      : Kn + (((size_t)b * CCH + h * HD + (d - HD)) << 16);
  const int n0 = chunk * CHUNKLEN;
  v8f acc = {};
  for (int k = n0 + 2 * half; k < n0 + CHUNKLEN; k += 4) {
    v2f a;  a.x  = qrow[k]; a.y  = qrow[k + 1];
    v2f bb; bb.x = krow[k]; bb.y = krow[k + 1];
    acc = __builtin_amdgcn_wmma_f32_16x16x4_f32(false, a, false, bb, (short)0, acc, false, false);
  }
#pragma unroll
  for (int v = 0; v < 8; ++v) {
    const int c = mt * 16 + half * 8 + v;
    P[(((size_t)bh * HD + c) * 64 + d) * NCHUNK + chunk] = acc[v];
  }
}

__global__ __launch_bounds__(256) void chunksum_temp(
    const float* __restrict__ P, const float* __restrict__ temp, float* __restrict__ L) {
  const int idx = blockIdx.x * 256 + threadIdx.x;   // < 12*32*64 = 24576
  const float* p = P + (size_t)idx * NCHUNK;
  float s = 0.f;
  for (int i = 0; i < NCHUNK; ++i) s += p[i];
  const int h = (idx >> 11) % HEADS;                // idx = ((b*6+h)*32+c)*64+d
  L[idx] = s * temp[h];
}

// warp softmax over 64 entries; one wave per (bh,c) row (384 rows)
__global__ __launch_bounds__(256) void softmax64(float* __restrict__ L) {
  const int wid = blockIdx.x * 8 + (threadIdx.x >> 5);   // 0..383
  const int lane = threadIdx.x & 31;
  float* row = L + (size_t)wid * 64;
  float a = row[lane], b = row[lane + 32];
  float m = fmaxf(a, b);
#pragma unroll
  for (int off = 16; off; off >>= 1) m = fmaxf(m, __shfl_xor(m, off, 32));
  float ea = __expf(a - m), eb = __expf(b - m);
  float s = ea + eb;
#pragma unroll
  for (int off = 16; off; off >>= 1) s += __shfl_xor(s, off, 32);
  const float r = 1.0f / s;
  row[lane] = ea * r; row[lane + 32] = eb * r;
}

// ---------------------------------------------------------------------------
// out[bh,c,n] = sum_d attn[bh,c,d] * Vfull[d,n]; K=64, wave -> 32x64 tile
// (both 16-row m-tiles of the head share the B fragments). grid = (128, 1, 12)
// ---------------------------------------------------------------------------
__global__ __launch_bounds__(256) void attn_v_wmma(
    const float* __restrict__ A, const float* __restrict__ Vc,
    const float* __restrict__ Vn, float* __restrict__ O) {
  const int lane = threadIdx.x & 31;
  const int wave = threadIdx.x >> 5;
  const int l16  = lane & 15;
  const int half = lane >> 4;
  const int bh = blockIdx.z;
  const int n0 = (blockIdx.x * 8 + wave) * 64;
  const int b = bh / HEADS, h = bh % HEADS;
  const float* arow0 = A + ((size_t)bh * HD + l16) * 64;        // rows 0..15
  const float* arow1 = arow0 + (size_t)16 * 64;                  // rows 16..31
  v8f accA0 = {}, accA1 = {}, accA2 = {}, accA3 = {};
  v8f accB0 = {}, accB1 = {}, accB2 = {}, accB3 = {};
  for (int k = 0; k < 64; k += 4) {
    const int d = k + 2 * half;               // even; d and d+1 never straddle 31/32
    v2f a0; a0.x = arow0[d]; a0.y = arow0[d + 1];
    v2f a1; a1.x = arow1[d]; a1.y = arow1[d + 1];
    const float* v0 = (d < HD) ? Vc + (((size_t)bh * HD + d) << 16)
                               : Vn + (((size_t)b * CCH + h * HD + (d - HD)) << 16);
    const float* v1 = v0 + HW;                // row d+1 in same buffer
    const int nb = n0 + l16;
    v2f b0; b0.x = v0[nb + 0];  b0.y = v1[nb + 0];
    v2f b1; b1.x = v0[nb + 16]; b1.y = v1[nb + 16];
    v2f b2; b2.x = v0[nb + 32]; b2.y = v1[nb + 32];
    v2f b3; b3.x = v0[nb + 48]; b3.y = v1[nb + 48];
    accA0 = __builtin_amdgcn_wmma_f32_16x16x4_f32(false, a0, false, b0, (short)0, accA0, false, false);
    accA1 = __builtin_amdgcn_wmma_f32_16x16x4_f32(false, a0, false, b1, (short)0, accA1, false, false);
    accA2 = __builtin_amdgcn_wmma_f32_16x16x4_f32(false, a0, false, b2, (short)0, accA2, false, false);
    accA3 = __builtin_amdgcn_wmma_f32_16x16x4_f32(false, a0, false, b3, (short)0, accA3, false, false);
    accB0 = __builtin_amdgcn_wmma_f32_16x16x4_f32(false, a1, false, b0, (short)0, accB0, false, false);
    accB1 = __builtin_amdgcn_wmma_f32_16x16x4_f32(false, a1, false, b1, (short)0, accB1, false, false);
    accB2 = __builtin_amdgcn_wmma_f32_16x16x4_f32(false, a1, false, b2, (short)0, accB2, false, false);
    accB3 = __builtin_amdgcn_wmma_f32_16x16x4_f32(false, a1, false, b3, (short)0, accB3, false, false);
  }
  float* o0 = O + (((size_t)b * CCH + h * HD + half * 8) << 16) + n0 + l16;
  float* o1 = o0 + ((size_t)16 << 16);
#pragma unroll
  for (int v = 0; v < 8; ++v) {
    o0[((size_t)v << 16) + 0]  = accA0[v];
    o0[((size_t)v << 16) + 16] = accA1[v];
    o0[((size_t)v << 16) + 32] = accA2[v];
    o0[((size_t)v << 16) + 48] = accA3[v];
    o1[((size_t)v << 16) + 0]  = accB0[v];
    o1[((size_t)v << 16) + 16] = accB1[v];
    o1[((size_t)v << 16) + 32] = accB2[v];
    o1[((size_t)v << 16) + 48] = accB3[v];
  }
}

// ---------------------------------------------------------------------------
extern "C" void kernel_launch(void* const* d_in, const int* in_sizes, int n_in,
                              void* d_out, int out_size, void* d_ws, size_t ws_size,
                              hipStream_t stream) {
  const float* x     = (const float*)d_in[0];   // [2,192,256,256]
  const float* kc    = (const float*)d_in[1];   // [2,6,32,65536]
  const float* vc    = (const float*)d_in[2];   // [2,6,32,65536]
  const float* wqkv  = (const float*)d_in[3];   // [576,192]
  const float* wdw   = (const float*)d_in[4];   // [576,1,3,3]
  const float* wproj = (const float*)d_in[5];   // [192,192]
  const float* temp  = (const float*)d_in[6];   // [6]

  const size_t NOUT = (size_t)BATCH * CCH * HW;        // 25,165,824 floats per tensor
  float* out_y = (float*)d_out;                        // [2,192,65536]
  float* out_k = out_y + NOUT;                         // normalized new k
  float* out_v = out_k + NOUT;                         // new v

  float* ws = (float*)d_ws;
  float* qkv_lin = ws;                                 // [2,576,65536] (reused as out_attn)
  float* qbuf    = qkv_lin + (size_t)BATCH * C3 * HW;  // [2,192,65536]
  float* part    = qbuf + NOUT;                        // [24576,128]
  float* logits  = part + (size_t)12 * HD * 64 * NCHUNK; // [24576]
  float* rinv    = logits + (size_t)12 * HD * 64;      // [768]
  float* out_att = qkv_lin;                            // alias: qkv_lin dead after dwconv

  // 1. qkv = conv1x1(x, w_qkv)
  gemm32_wmma<<<dim3(HW / 512, C3 / 32, BATCH), 256, 0, stream>>>(
      wqkv, x, qkv_lin, C3, CCH, HW);
  // 2. depthwise 3x3 -> q(ws), k(raw->d_out), v(d_out)
  dwconv3x3_split<<<(unsigned)((size_t)BATCH * C3 * HW / 256), 256, 0, stream>>>(
      qkv_lin, wdw, qbuf, out_k, out_v);
  // 3. row norms for q and k
  rownorm<<<2 * BATCH * CCH, 256, 0, stream>>>(qbuf, out_k, rinv);
  // 4. apply 1/norm in place
  scale_rows<<<(unsigned)(2 * NOUT / 256), 256, 0, stream>>>(qbuf, out_k, rinv);
  // 5. split-K q.k^T partials
  attn_qk_partial<<<dim3(NCHUNK / 8, 8, BATCH * HEADS), 256, 0, stream>>>(
      qbuf, kc, out_k, part);
  // 6. reduce chunks * temperature
  chunksum_temp<<<(12 * HD * 64) / 256, 256, 0, stream>>>(part, temp, logits);
  // 7. softmax over 64
  softmax64<<<(BATCH * HEADS * HD) / 8, 256, 0, stream>>>(logits);
  // 8. attn @ Vfull (32x64 tiles per wave)
  attn_v_wmma<<<dim3(HW / 512, 1, BATCH * HEADS), 256, 0, stream>>>(
      logits, vc, out_v, out_att);
  // 9. out = conv1x1(out_att, w_proj)
  gemm32_wmma<<<dim3(HW / 512, CCH / 32, BATCH), 256, 0, stream>>>(
      wproj, out_att, out_y, CCH, CCH, HW);
}